// EdgePredictionNet_55937654063333
// MI455X (gfx1250) — compile-verified
//
#include <hip/hip_runtime.h>

typedef __attribute__((ext_vector_type(2))) float v2f;
typedef __attribute__((ext_vector_type(8))) float v8f;

#define N_NODES 50000
#define N_EDGES 800000
#define N_LABEL 200000

// ---------------------------------------------------------------------------
// Degree: deg[t] += 1 for each edge
// ---------------------------------------------------------------------------
__global__ void deg_kernel(const int* __restrict__ tgt, float* __restrict__ deg) {
    int e = blockIdx.x * blockDim.x + threadIdx.x;
    if (e < N_EDGES) atomicAdd(&deg[tgt[e]], 1.0f);
}

// ---------------------------------------------------------------------------
// Scatter-add: agg[tgt[e]] += x[src[e]]  (one thread per (edge, float4-chunk))
// ---------------------------------------------------------------------------
template <int DIN>
__global__ void scatter_kernel(const float* __restrict__ x,
                               const int* __restrict__ src,
                               const int* __restrict__ tgt,
                               float* __restrict__ agg) {
    constexpr int CHUNKS = DIN / 4;
    long long idx = (long long)blockIdx.x * blockDim.x + threadIdx.x;
    int e = (int)(idx / CHUNKS);
    int c = (int)(idx % CHUNKS);
    if (e >= N_EDGES) return;
    int s = src[e];
    int t = tgt[e];
    const float4 v = ((const float4*)(x + (size_t)s * DIN))[c];
    float* a = agg + (size_t)t * DIN + c * 4;
    atomicAdd(a + 0, v.x);
    atomicAdd(a + 1, v.y);
    atomicAdd(a + 2, v.z);
    atomicAdd(a + 3, v.w);
}

// ---------------------------------------------------------------------------
// Fused SAGE layer: out = act( (agg/deg) @ Wl + x @ Wr + b )
// One wave computes a 16x16 output tile via V_WMMA_F32_16X16X4_F32.
// Grid must be sized so every wave owns a valid tile (EXEC all-ones for WMMA).
// ---------------------------------------------------------------------------
template <int DIN, int DOUT, bool RELU>
__global__ void sage_gemm_wmma(const float* __restrict__ agg,
                               const float* __restrict__ x,
                               const float* __restrict__ deg,
                               const float* __restrict__ Wl,
                               const float* __restrict__ Wr,
                               const float* __restrict__ bias,
                               float* __restrict__ out) {
    const int lane  = threadIdx.x & 31;
    const int wave  = threadIdx.x >> 5;
    const int wpb   = blockDim.x >> 5;
    const int tile  = blockIdx.x * wpb + wave;
    constexpr int TILES_N = DOUT / 16;
    const int tm = tile / TILES_N;   // node-row tile  (0..3124)
    const int tn = tile % TILES_N;   // out-col tile

    const int half = lane >> 4;      // 0: lanes 0-15, 1: lanes 16-31
    const int l15  = lane & 15;
    const int rowA = tm * 16 + l15;  // A row this lane feeds
    const int col  = tn * 16 + l15;  // B/C/D column this lane holds

    const float inv = 1.0f / fmaxf(deg[rowA], 1.0f);

    const float* __restrict__ aggRow = agg + (size_t)rowA * DIN;
    const float* __restrict__ xRow   = x   + (size_t)rowA * DIN;

    v8f acc = {0.f, 0.f, 0.f, 0.f, 0.f, 0.f, 0.f, 0.f};

    // mean @ Wl  (deg-normalization folded into A loads)
#pragma unroll 4
    for (int k0 = 0; k0 < DIN; k0 += 4) {
        const int kk = k0 + 2 * half;
        v2f a, b;
        a.x = aggRow[kk]     * inv;
        a.y = aggRow[kk + 1] * inv;
        b.x = Wl[(size_t)kk       * DOUT + col];
        b.y = Wl[(size_t)(kk + 1) * DOUT + col];
        acc = __builtin_amdgcn_wmma_f32_16x16x4_f32(
            /*neg_a=*/false, a, /*neg_b=*/false, b,
            /*c_mod=*/(short)0, acc, /*reuse_a=*/false, /*reuse_b=*/false);
    }
    // x @ Wr  (same accumulator)
#pragma unroll 4
    for (int k0 = 0; k0 < DIN; k0 += 4) {
        const int kk = k0 + 2 * half;
        v2f a, b;
        a.x = xRow[kk];
        a.y = xRow[kk + 1];
        b.x = Wr[(size_t)kk       * DOUT + col];
        b.y = Wr[(size_t)(kk + 1) * DOUT + col];
        acc = __builtin_amdgcn_wmma_f32_16x16x4_f32(
            false, a, false, b, (short)0, acc, false, false);
    }

    const float bv = bias[col];
#pragma unroll
    for (int r = 0; r < 8; ++r) {
        float v = acc[r] + bv;
        if (RELU) v = fmaxf(v, 0.0f);
        const int m = tm * 16 + r + 8 * half;   // C/D layout: VGPR r -> row r / r+8
        out[(size_t)m * DOUT + col] = v;
    }
}

// ---------------------------------------------------------------------------
// Decode: out[e] = dot(z[s], z[t]) over 64 dims — one wave32 per label edge
// ---------------------------------------------------------------------------
__global__ void decode_kernel(const float* __restrict__ z,
                              const int* __restrict__ ls,
                              const int* __restrict__ lt,
                              float* __restrict__ out) {
    const int lane = threadIdx.x & 31;
    const long long wave = ((long long)blockIdx.x * blockDim.x + threadIdx.x) >> 5;
    if (wave >= N_LABEL) return;
    const int s = ls[wave];
    const int t = lt[wave];
    const float* zs = z + (size_t)s * 64;
    const float* zt = z + (size_t)t * 64;
    float v = zs[lane] * zt[lane] + zs[lane + 32] * zt[lane + 32];
#pragma unroll
    for (int off = 16; off >= 1; off >>= 1)
        v += __shfl_xor(v, off, 32);
    if (lane == 0) out[wave] = v;
}

// ---------------------------------------------------------------------------
// Host-side orchestration (all async on `stream`, graph-capture safe)
// ---------------------------------------------------------------------------
extern "C" void kernel_launch(void* const* d_in, const int* in_sizes, int n_in,
                              void* d_out, int out_size, void* d_ws, size_t ws_size,
                              hipStream_t stream) {
    const float* x    = (const float*)d_in[0];
    const int*   eidx = (const int*)d_in[1];   // [2, N_EDGES]
    const int*   lidx = (const int*)d_in[2];   // [2, N_LABEL]
    const float* Wl1 = (const float*)d_in[3];
    const float* Wr1 = (const float*)d_in[4];
    const float* b1  = (const float*)d_in[5];
    const float* Wl2 = (const float*)d_in[6];
    const float* Wr2 = (const float*)d_in[7];
    const float* b2  = (const float*)d_in[8];
    const float* Wl3 = (const float*)d_in[9];
    const float* Wr3 = (const float*)d_in[10];
    const float* b3  = (const float*)d_in[11];

    const int* src = eidx;
    const int* tgt = eidx + N_EDGES;
    const int* ls  = lidx;
    const int* lt  = lidx + N_LABEL;

    // Workspace carve-out (floats): deg[N], agg[N*128], h1[N*128], h2[N*128], z[N*64]
    float* deg = (float*)d_ws;
    float* agg = deg + N_NODES;
    float* h1  = agg + (size_t)N_NODES * 128;
    float* h2  = h1  + (size_t)N_NODES * 128;
    float* z   = h2  + (size_t)N_NODES * 128;
    float* outp = (float*)d_out;

    // Degree (computed fresh every call — deterministic pipeline)
    hipMemsetAsync(deg, 0, (size_t)N_NODES * sizeof(float), stream);
    deg_kernel<<<(N_EDGES + 255) / 256, 256, 0, stream>>>(tgt, deg);

    // ---- Layer 1: 64 -> 128, ReLU ----
    hipMemsetAsync(agg, 0, (size_t)N_NODES * 64 * sizeof(float), stream);
    {
        long long threads = (long long)N_EDGES * (64 / 4);
        scatter_kernel<64><<<(int)((threads + 255) / 256), 256, 0, stream>>>(x, src, tgt, agg);
    }
    // 3125 row-tiles * 8 col-tiles = 25000 tiles; 8 waves/block -> 3125 blocks exact
    sage_gemm_wmma<64, 128, true><<<3125, 256, 0, stream>>>(agg, x, deg, Wl1, Wr1, b1, h1);

    // ---- Layer 2: 128 -> 128, ReLU ----
    hipMemsetAsync(agg, 0, (size_t)N_NODES * 128 * sizeof(float), stream);
    {
        long long threads = (long long)N_EDGES * (128 / 4);
        scatter_kernel<128><<<(int)((threads + 255) / 256), 256, 0, stream>>>(h1, src, tgt, agg);
    }
    sage_gemm_wmma<128, 128, true><<<3125, 256, 0, stream>>>(agg, h1, deg, Wl2, Wr2, b2, h2);

    // ---- Layer 3: 128 -> 64, no activation ----
    hipMemsetAsync(agg, 0, (size_t)N_NODES * 128 * sizeof(float), stream);
    {
        long long threads = (long long)N_EDGES * (128 / 4);
        scatter_kernel<128><<<(int)((threads + 255) / 256), 256, 0, stream>>>(h2, src, tgt, agg);
    }
    // 3125 row-tiles * 4 col-tiles = 12500 tiles; 4 waves/block -> 3125 blocks exact
    sage_gemm_wmma<128, 64, false><<<3125, 128, 0, stream>>>(agg, h2, deg, Wl3, Wr3, b3, z);

    // ---- Decode: one wave per label edge; 8 waves/block -> 25000 blocks exact ----
    decode_kernel<<<N_LABEL / 8, 256, 0, stream>>>(z, ls, lt, outp);
}